// QueryAndGroup_21715354649168
// MI455X (gfx1250) — compile-verified
//
#include <hip/hip_runtime.h>

// Problem constants (from setup_inputs / reference)
#define BATCH 4
#define NPTS  16384      // N
#define NCTR  4096       // P
#define NSAMP 32         // nsample
#define NFEAT 64         // feature channels
#define CTOT  (3 + NFEAT)
#define RAD2  0.04f      // 0.2^2

typedef float v2f __attribute__((ext_vector_type(2)));
typedef float v8f __attribute__((ext_vector_type(8)));

// ---------------------------------------------------------------------------
// Kernel 1: ball query via V_WMMA_F32_16X16X4_F32 (fp32, K=4 -> K=3 geometry).
// One wave32 per 16-center tile, 32 points per loop trip (2 WMMAs).
//   D[m][n] = A(16x4) x B(4x16) + C(16x16)
//   A[m] = [-2cx, -2cy, -2cz, 1], B[:,n] = [x, y, z, |x|^2], C[m][n] = |c_m|^2
//   => D[m][n] = dist^2(center m, point n); compare vs RAD2.
// Layouts per ISA 7.12.2 (32-bit A 16x4: lanes 0-15 K0/K1, lanes 16-31 K2/K3;
// C/D: VGPR r = row r / r+8 per half-wave, col = lane%16; B mirrors A).
// Selection: __ballot per D-register -> 16 hit bits per center; the two
// chunks' masks concatenate into a 32-bit pend per owner lane (lane = center),
// popped in index order => exact CUDA first-32 ball_query semantics.
// ---------------------------------------------------------------------------
__global__ __launch_bounds__(32)
void ball_query_wmma(const float* __restrict__ xyz,
                     const float* __restrict__ new_xyz,
                     int* __restrict__ idx_out)
{
    const int lane = threadIdx.x;                 // 0..31 (wave32)
    const int tile = blockIdx.x;                  // BATCH*NCTR/16 tiles
    const int b    = tile / (NCTR / 16);
    const int pb   = (tile % (NCTR / 16)) * 16;   // first center of tile
    const int m    = lane & 15;

    // Center for row m (both half-waves load the same 16 centers)
    const float* cp = new_xyz + ((size_t)b * NCTR + pb + m) * 3;
    const float cx = cp[0], cy = cp[1], cz = cp[2];
    const float cn2 = cx * cx + cy * cy + cz * cz;

    // A matrix registers
    v2f a;
    a[0] = (lane < 16) ? (-2.0f * cx) : (-2.0f * cz);
    a[1] = (lane < 16) ? (-2.0f * cy) : 1.0f;

    // C accumulator: C[m][n] = |c_m|^2 ; lane holds rows (r + 8*halfwave)
    v8f cacc;
    const int half = (lane >= 16) ? 8 : 0;
#pragma unroll
    for (int r = 0; r < 8; ++r)
        cacc[r] = __shfl(cn2, r + half, 32);

    int cnt = 0, first = 0;
    int* myidx = idx_out + ((size_t)b * NCTR + pb + m) * NSAMP;

    const float* base = xyz + (size_t)b * NPTS * 3;

    for (int n0 = 0; n0 < NPTS; n0 += 32) {
        // Issue both 16-point chunk loads back-to-back (one loadcnt wait),
        // and prefetch the next 32-point chunk into WGP$/L2.
        const float* pp0 = base + (size_t)(n0 + m) * 3;
        const float* pp1 = pp0 + 16 * 3;
        __builtin_prefetch(pp0 + 32 * 3, 0, 3);
        const float x0 = pp0[0], y0 = pp0[1], z0 = pp0[2];
        const float x1 = pp1[0], y1 = pp1[1], z1 = pp1[2];

        const float n20 = x0 * x0 + y0 * y0 + z0 * z0;
        const float n21 = x1 * x1 + y1 * y1 + z1 * z1;

        v2f b0, b1;
        b0[0] = (lane < 16) ? x0 : z0;
        b0[1] = (lane < 16) ? y0 : n20;
        b1[0] = (lane < 16) ? x1 : z1;
        b1[1] = (lane < 16) ? y1 : n21;

        // Two d2 tiles: 16 centers x 32 points
        v8f d0 = __builtin_amdgcn_wmma_f32_16x16x4_f32(
            false, a, false, b0, (short)0, cacc, false, false);
        v8f d1 = __builtin_amdgcn_wmma_f32_16x16x4_f32(
            false, a, false, b1, (short)0, cacc, false, false);

        // Per-center 32-bit hit mask (bit k <-> point n0+k)
        unsigned pend = 0;
#pragma unroll
        for (int r = 0; r < 8; ++r) {
            unsigned lo = (unsigned)__ballot(d0[r] < RAD2);
            if (lane == r)     pend |= lo & 0xffffu;
            if (lane == r + 8) pend |= (lo >> 16) & 0xffffu;
        }
#pragma unroll
        for (int r = 0; r < 8; ++r) {
            unsigned hi = (unsigned)__ballot(d1[r] < RAD2);
            if (lane == r)     pend |= (hi & 0xffffu) << 16;
            if (lane == r + 8) pend |= ((hi >> 16) & 0xffffu) << 16;
        }

        // Owner lanes append hits in index order (first-32 semantics)
        if (lane < 16) {
            while (pend && cnt < NSAMP) {
                const int t = __ffs(pend) - 1;
                pend &= pend - 1;
                const int gi = n0 + t;
                if (cnt == 0) first = gi;
                myidx[cnt] = gi;
                ++cnt;
            }
        }

        // Uniform early exit once every center has 32 hits
        const int done = (lane < 16) ? (cnt >= NSAMP) : 1;
        if (__all(done)) break;
    }

    // Pad remaining slots with first hit (0 if none found: first inits to 0)
    if (lane < 16) {
        for (int k = cnt; k < NSAMP; ++k) myidx[k] = first;
    }
}

// ---------------------------------------------------------------------------
// Kernel 2: gather + concat (memory-bound, ~137 MB output stream).
// One 256-thread block per (b,p): idx+center staged in LDS, then 67x32
// elements written coalesced. Channels 0..2 = xyz[idx]-center, 3..66 = feat.
// Feature table is 16 MB total with heavy reuse -> L2-resident on 192 MB L2.
// ---------------------------------------------------------------------------
__global__ __launch_bounds__(256)
void group_concat(const float* __restrict__ xyz,
                  const float* __restrict__ new_xyz,
                  const float* __restrict__ feat,
                  const int* __restrict__ idx,
                  float* __restrict__ out)
{
    const int bp = blockIdx.x;            // b*NCTR + p
    const int b  = bp / NCTR;
    const int p  = bp % NCTR;
    const int tid = threadIdx.x;

    __shared__ int   sidx[NSAMP];
    __shared__ float scen[3];
    if (tid < NSAMP) sidx[tid] = idx[(size_t)bp * NSAMP + tid];
    if (tid < 3)     scen[tid] = new_xyz[(size_t)bp * 3 + tid];
    __syncthreads();

    for (int e = tid; e < CTOT * NSAMP; e += 256) {
        const int c = e >> 5;        // channel 0..66
        const int s = e & 31;        // sample  0..31
        const int gi = sidx[s];
        float v;
        if (c < 3) v = xyz[((size_t)b * NPTS + gi) * 3 + c] - scen[c];
        else       v = feat[((size_t)b * NFEAT + (c - 3)) * NPTS + gi];
        out[(((size_t)b * CTOT + c) * NCTR + p) * NSAMP + s] = v;
    }
}

extern "C" void kernel_launch(void* const* d_in, const int* in_sizes, int n_in,
                              void* d_out, int out_size, void* d_ws, size_t ws_size,
                              hipStream_t stream) {
    const float* xyz  = (const float*)d_in[0];   // (4,16384,3)
    const float* nxyz = (const float*)d_in[1];   // (4,4096,3)
    const float* feat = (const float*)d_in[2];   // (4,64,16384)
    int*   idx = (int*)d_ws;                     // (4,4096,32) = 2 MB scratch
    float* out = (float*)d_out;                  // (4,67,4096,32)

    ball_query_wmma<<<BATCH * NCTR / 16, 32, 0, stream>>>(xyz, nxyz, idx);
    group_concat<<<BATCH * NCTR, 256, 0, stream>>>(xyz, nxyz, feat, idx, out);
}